// DRAELoss_46024869544164
// MI455X (gfx1250) — compile-verified
//
#include <hip/hip_runtime.h>

typedef __attribute__((ext_vector_type(2))) float v2f;
typedef __attribute__((ext_vector_type(8))) float v8f;

#define NROWS 8192
#define NCOLS 4096
#define ROWS_PER_BLK 16   // 256 threads: 16 lanes per row
#define LAMB 0.1f

// ---------------------------------------------------------------------------
// Kernel 1: per-row sum of squared differences.
// 512 blocks x 256 threads; each block owns 16 rows; each row has 16 lanes,
// each lane accumulates 64 float4's (coalesced: 16 lanes * 16B = 256B bursts).
// Final 16x16 partial reduction done on the matrix pipe with
// V_WMMA_F32_16X16X4_F32 (A = partials, B = ones, C accumulates).
// ---------------------------------------------------------------------------
__global__ __launch_bounds__(256) void drae_rowsumsq(const float* __restrict__ inp,
                                                     const float* __restrict__ tgt,
                                                     float* __restrict__ Err) {
  __shared__ float part[ROWS_PER_BLK][16];
  const int tid = threadIdx.x;
  const int rowInBlk = tid >> 4;
  const int lane16 = tid & 15;
  const int row = blockIdx.x * ROWS_PER_BLK + rowInBlk;

  const float4* pa = (const float4*)(inp + (size_t)row * NCOLS);
  const float4* pb = (const float4*)(tgt + (size_t)row * NCOLS);

  float acc = 0.f;
  const int iters = (NCOLS / 4) / 16; // 64
  #pragma unroll 4
  for (int it = 0; it < iters; ++it) {
    const int idx = it * 16 + lane16;
    if ((it & 15) == 0) {
      // speculative stream prefetch (global_prefetch_b8); OOB is silently dropped
      __builtin_prefetch((const void*)(pa + idx + 256), 0, 0);
      __builtin_prefetch((const void*)(pb + idx + 256), 0, 0);
    }
    const float4 a = pa[idx];
    const float4 b = pb[idx];
    const float dx = a.x - b.x, dy = a.y - b.y, dz = a.z - b.z, dw = a.w - b.w;
    acc += dx * dx + dy * dy + dz * dz + dw * dw;
  }
  part[rowInBlk][lane16] = acc;
  __syncthreads();

  // Wave 0 (threads 0..31, EXEC all ones) reduces part[16][16] -> 16 row sums
  // with 4 accumulating V_WMMA_F32_16X16X4_F32 (B = ones).
  if (tid < 32) {
    const int l = tid;
    const int m = l & 15;             // A-matrix row M (both lane halves)
    const int koff = (l >> 4) * 2;    // lanes 0-15: K=0,1 ; lanes 16-31: K=2,3
    v2f bv; bv[0] = 1.0f; bv[1] = 1.0f;
    v8f c = {};
    #pragma unroll
    for (int c4 = 0; c4 < 4; ++c4) {
      v2f av;
      av[0] = part[m][c4 * 4 + koff + 0];
      av[1] = part[m][c4 * 4 + koff + 1];
      c = __builtin_amdgcn_wmma_f32_16x16x4_f32(
          /*neg_a=*/false, av, /*neg_b=*/false, bv,
          /*c_mod=*/(short)0, c, /*reuse_a=*/false, /*reuse_b=*/false);
    }
    // C/D layout: lanes 0-15 hold rows 0..7 in vgpr 0..7 (replicated over N),
    // lanes 16-31 hold rows 8..15. Extract via lane 0 and lane 16.
    const int mbase = (l >> 4) * 8;
    if ((l & 15) == 0) {
      #pragma unroll
      for (int i = 0; i < 8; ++i)
        Err[blockIdx.x * ROWS_PER_BLK + mbase + i] = c[i];
    }
  }
}

// ---------------------------------------------------------------------------
// Kernel 2: single 1024-thread workgroup.
// Bitonic sort of 8192 values in LDS, two-level inclusive scans of s and s^2,
// objective evaluation, argmin (first-occurrence), final scalar.
// ---------------------------------------------------------------------------
__global__ __launch_bounds__(1024) void drae_phase2(const float* __restrict__ Err,
                                                    float* __restrict__ out) {
  __shared__ float s[NROWS];       // 32 KB
  __shared__ float scanS[1024];    // 4 KB
  __shared__ float scanS2[1024];   // 4 KB
  __shared__ float redObj[1024];   // 4 KB
  __shared__ int   redJ[1024];     // 4 KB
  __shared__ float redCs[1024];    // 4 KB
  const int tid = threadIdx.x;

  for (int i = tid; i < NROWS; i += 1024) s[i] = Err[i];
  __syncthreads();

  // Bitonic sort, ascending.
  for (unsigned size = 2; size <= (unsigned)NROWS; size <<= 1) {
    for (unsigned stride = size >> 1; stride > 0; stride >>= 1) {
      for (unsigned i = tid; i < (unsigned)NROWS; i += 1024) {
        const unsigned j = i ^ stride;
        if (j > i) {
          const bool up = ((i & size) == 0);
          const float a = s[i], b = s[j];
          if ((a > b) == up) { s[i] = b; s[j] = a; }
        }
      }
      __syncthreads();
    }
  }

  // Per-thread serial inclusive scan over its chunk of 8 sorted values.
  float ls[8], ls2[8];
  const int base = tid * 8;
  float sumS = 0.f, sumS2 = 0.f;
  #pragma unroll
  for (int r = 0; r < 8; ++r) {
    const float v = s[base + r];
    sumS += v;      ls[r]  = sumS;
    sumS2 += v * v; ls2[r] = sumS2;
  }
  scanS[tid] = sumS; scanS2[tid] = sumS2;
  __syncthreads();

  // Hillis-Steele inclusive scan over the 1024 chunk totals.
  for (int off = 1; off < 1024; off <<= 1) {
    const float a = scanS[tid], a2 = scanS2[tid];
    float b = 0.f, b2 = 0.f;
    if (tid >= off) { b = scanS[tid - off]; b2 = scanS2[tid - off]; }
    __syncthreads();
    scanS[tid] = a + b; scanS2[tid] = a2 + b2;
    __syncthreads();
  }

  const float total  = scanS[1023];
  const float total2 = scanS2[1023];
  const float exS  = tid ? scanS[tid - 1]  : 0.f;
  const float exS2 = tid ? scanS2[tid - 1] : 0.f;

  const float nf = (float)NROWS;
  const float allMean = total / nf;
  const float Sb = total2 - nf * allMean * allMean;

  // Objective over split indices j (0-based obj index; k = j+1), j in [0, N-2].
  float bestObj = __builtin_inff();
  int bestJ = 0; float bestCs = 0.f;
  #pragma unroll
  for (int r = 0; r < 8; ++r) {
    const int j = base + r;
    if (j >= NROWS - 1) break;
    const float csk  = exS  + ls[r];   // inclusive cs at index j
    const float cs2k = exS2 + ls2[r];  // inclusive cs2 at index j
    const float kf = (float)(j + 1);
    const float Sw1 = cs2k - csk * csk / kf;
    const float rest = total - csk;
    const float Sw2 = total2 - cs2k - rest * rest / (nf - kf);
    const float obj = (Sw1 + Sw2) / Sb;
    if (obj < bestObj) { bestObj = obj; bestJ = j; bestCs = csk; }  // first min wins
  }
  redObj[tid] = bestObj; redJ[tid] = bestJ; redCs[tid] = bestCs;
  __syncthreads();

  // Argmin tree reduction; lower index wins ties (higher tid always has larger j).
  for (int st = 512; st > 0; st >>= 1) {
    if (tid < st) {
      const float o2 = redObj[tid + st];
      if (o2 < redObj[tid]) {
        redObj[tid] = o2; redJ[tid] = redJ[tid + st]; redCs[tid] = redCs[tid + st];
      }
    }
    __syncthreads();
  }

  if (tid == 0) {
    const float T = (float)(redJ[0] + 1);
    out[0] = redCs[0] / T + LAMB * redObj[0];   // size_average=True
  }
}

extern "C" void kernel_launch(void* const* d_in, const int* in_sizes, int n_in,
                              void* d_out, int out_size, void* d_ws, size_t ws_size,
                              hipStream_t stream) {
  const float* inp = (const float*)d_in[0];
  const float* tgt = (const float*)d_in[1];
  float* outp = (float*)d_out;
  float* Err = (float*)d_ws;  // 8192 floats of scratch

  const int nblocks = NROWS / ROWS_PER_BLK;  // 512
  drae_rowsumsq<<<dim3(nblocks), dim3(256), 0, stream>>>(inp, tgt, Err);
  drae_phase2<<<dim3(1), dim3(1024), 0, stream>>>(Err, outp);
}